// ModelClass_70205535420832
// MI455X (gfx1250) — compile-verified
//
#include <hip/hip_runtime.h>

// ---------------------------------------------------------------------------
// GIN-style GNN forward for MI455X (gfx1250), fp32 end-to-end.
// GEMMs use native fp32 WMMA: v_wmma_f32_16x16x4_f32 (wave32, 16x16 C tile).
// Each wave computes a 32x64 output tile (2 M-tiles sharing B fragments):
// per k-step 4 ds_load_b64 + 2 global_load_b64 feed 8 WMMAs.
// ---------------------------------------------------------------------------

typedef float v2f __attribute__((ext_vector_type(2)));
typedef float v8f __attribute__((ext_vector_type(8)));

#define NN 50000      // nodes
#define NE 800000     // edges
#define NG 128        // graphs
#define NDYN 64
#define NALL 72       // 64 dyn + 4 static + 4 hlv
#define DEEP 256

// ---------------------------------------------------------------------------
// small elementwise kernels
// ---------------------------------------------------------------------------
__global__ void zero_f32(float* __restrict__ p, int n) {
  int i = blockIdx.x * blockDim.x + threadIdx.x;
  if (i < n) p[i] = 0.0f;
}

__global__ void init_xd(const float* __restrict__ x, float* __restrict__ xd) {
  int i = blockIdx.x * blockDim.x + threadIdx.x;
  if (i >= NN * NDYN) return;
  int node = i >> 6, c = i & 63;
  xd[i] = (c == 0) ? x[node] : 0.0f;
}

// h = concat(xd[:,0:64], static[:,0:4], hlvs[batch_idx][:,0:4])
__global__ void build_h(const float* __restrict__ xd, const float* __restrict__ stat,
                        const float* __restrict__ hlvs, const int* __restrict__ batch,
                        float* __restrict__ h) {
  int i = blockIdx.x * blockDim.x + threadIdx.x;
  if (i >= NN * NALL) return;
  int node = i / NALL;
  int c = i - node * NALL;
  float v;
  if (c < 64)      v = xd[node * 64 + c];
  else if (c < 68) v = stat[node * 4 + (c - 64)];
  else             v = hlvs[batch[node] * 4 + (c - 68)];
  h[i] = v;
}

// agg[dst] += h[src] over all edges (72 floats / edge)
__global__ void edge_agg(const float* __restrict__ h, const int* __restrict__ src,
                         const int* __restrict__ dst, float* __restrict__ agg) {
  int i = blockIdx.x * blockDim.x + threadIdx.x;
  if (i >= NE * NALL) return;
  int e = i / NALL;
  int c = i - e * NALL;
  atomicAdd(&agg[(size_t)dst[e] * NALL + c], h[(size_t)src[e] * NALL + c]);
}

// agg = (1+eps)*h + agg   (GIN pre-MLP input, computed in place in agg)
__global__ void finish_gin(const float* __restrict__ h, const float* __restrict__ eps,
                           float* __restrict__ agg) {
  int i = blockIdx.x * blockDim.x + threadIdx.x;
  if (i >= NN * NALL) return;
  agg[i] = (1.0f + eps[0]) * h[i] + agg[i];
}

// pooled[batch[i]] += xd[i]
__global__ void pool_sum(const float* __restrict__ xd, const int* __restrict__ batch,
                         float* __restrict__ pooled) {
  int i = blockIdx.x * blockDim.x + threadIdx.x;
  if (i >= NN * NDYN) return;
  int node = i >> 6, c = i & 63;
  atomicAdd(&pooled[batch[node] * NDYN + c], xd[i]);
}

// hin = concat(hlvs[g,0:4], pooled[g,0:64])  -> [128, 68]
__global__ void build_hin(const float* __restrict__ hlvs, const float* __restrict__ pooled,
                          float* __restrict__ hin) {
  int i = blockIdx.x * blockDim.x + threadIdx.x;
  if (i >= NG * 68) return;
  int g = i / 68, c = i - g * 68;
  hin[i] = (c < 4) ? hlvs[g * 4 + c] : pooled[g * NDYN + (c - 4)];
}

// out[g] = relu(dot(A[g,0:256], W[:,0]) + b[0])
__global__ void final_layer(const float* __restrict__ A, const float* __restrict__ W,
                            const float* __restrict__ b, float* __restrict__ out) {
  int g = blockIdx.x * blockDim.x + threadIdx.x;
  if (g >= NG) return;
  float s = b[0];
  #pragma unroll 8
  for (int j = 0; j < DEEP; ++j) s += A[g * DEEP + j] * W[j];
  out[g] = s > 0.0f ? s : 0.0f;
}

// ---------------------------------------------------------------------------
// WMMA GEMM core: NT M-tiles (16 rows each) x 64 columns per wave.
// B fragments (from LDS) are shared across the NT A fragments, so NT=2 doubles
// WMMA count per LDS load: 4 ds_load_b64 + NT global_load_b64 -> 4*NT WMMAs.
// ---------------------------------------------------------------------------
template <int NT>
__device__ __forceinline__ void gemm_core(const float* __restrict__ A,
                                          const float* __restrict__ lds,
                                          const float* __restrict__ bias,
                                          float* __restrict__ C,
                                          int K, int Kp, int Nf, int n0,
                                          int tile0, int lane) {
  const int lane16 = lane & 15;
  const int koff = (lane >> 4) << 1;   // 0 (lanes 0-15) or 2 (lanes 16-31)
  const int sw = lane16 << 2;

  v8f acc[NT][4];
  #pragma unroll
  for (int m = 0; m < NT; ++m)
    #pragma unroll
    for (int t = 0; t < 4; ++t) acc[m][t] = v8f{};

  const float* Arow[NT];
  #pragma unroll
  for (int m = 0; m < NT; ++m)
    Arow[m] = A + (size_t)((tile0 + m) * 16 + lane16) * K + koff;

  for (int k = 0; k < K; k += 4) {
    v2f a[NT];
    #pragma unroll
    for (int m = 0; m < NT; ++m) a[m] = *(const v2f*)(Arow[m] + k);
    const int kb = (k + koff) ^ sw;    // even -> b64-aligned LDS read
    #pragma unroll
    for (int t = 0; t < 4; ++t) {
      v2f b = *(const v2f*)&lds[(t * 16 + lane16) * Kp + kb];
      #pragma unroll
      for (int m = 0; m < NT; ++m)
        acc[m][t] = __builtin_amdgcn_wmma_f32_16x16x4_f32(
            false, a[m], false, b, (short)0, acc[m][t], false, false);
    }
  }

  // C/D layout: VGPR i -> (M = i, N = lane) lanes 0-15, (M = 8+i, N = lane-16)
  const int rsub = (lane >> 4) << 3;
  #pragma unroll
  for (int m = 0; m < NT; ++m) {
    const int rbase = (tile0 + m) * 16 + rsub;
    #pragma unroll
    for (int t = 0; t < 4; ++t) {
      const int col = n0 + t * 16 + lane16;
      const float bv = bias[col];
      #pragma unroll
      for (int i = 0; i < 8; ++i) {
        float v = acc[m][t][i] + bv;
        C[(size_t)(rbase + i) * Nf + col] = v > 0.0f ? v : 0.0f;
      }
    }
  }
}

// C[M,Nf] = relu(A[M,K] * W[K,Nf] + bias)
//   block = 256 threads = 8 waves; each wave -> 32x64 tile (2 x 16-row tiles).
//   grid  = (ceil(M/256), Nf/64). Requires M % 16 == 0, K % 4 == 0, Nf % 64 == 0.
//   W strip staged in LDS column-major with XOR swizzle (conflict-free b64 reads).
//   dynamic LDS = 64 * Kp * 4 bytes, Kp = roundup(K, 64)  (<= 64 KB for K <= 256)
__global__ __launch_bounds__(256)
void gemm_bias_relu(const float* __restrict__ A, const float* __restrict__ W,
                    const float* __restrict__ bias, float* __restrict__ C,
                    int M, int K, int Nf) {
  extern __shared__ float lds[];
  const int tid = threadIdx.x;
  const int n0 = blockIdx.y * 64;
  const int Kp = (K + 63) & ~63;

  // Stage W[0:K, n0:n0+64] into LDS, column-major with per-column XOR swizzle.
  // Swizzle s = 4*(c&15) permutes k within each aligned 64-block (bijective
  // because Kp is a multiple of 64), keeping even/odd pairs adjacent for b64.
  for (int i = tid; i < K * 64; i += 256) {
    int kk = i >> 6;        // 0..K-1
    int c  = i & 63;        // 0..63 (consecutive -> coalesced global reads)
    lds[c * Kp + (kk ^ ((c & 15) << 2))] = W[(size_t)kk * Nf + n0 + c];
  }
  __syncthreads();

  const int wave = tid >> 5;
  const int lane = tid & 31;
  const int tile0 = blockIdx.x * 16 + wave * 2;  // wave-uniform
  if (tile0 * 16 >= M) return;                   // EXEC stays all-1s per wave

  if ((tile0 + 1) * 16 < M + 15 && (tile0 + 1) * 16 <= M - 16 + 16 && (tile0 + 1) * 16 < M) {
    gemm_core<2>(A, lds, bias, C, K, Kp, Nf, n0, tile0, lane);
  } else {
    gemm_core<1>(A, lds, bias, C, K, Kp, Nf, n0, tile0, lane);
  }
}

// ---------------------------------------------------------------------------
// host side
// ---------------------------------------------------------------------------
static inline size_t align256(size_t x) { return (x + 255) & ~(size_t)255; }

extern "C" void kernel_launch(void* const* d_in, const int* in_sizes, int n_in,
                              void* d_out, int out_size, void* d_ws, size_t ws_size,
                              hipStream_t stream) {
  (void)in_sizes; (void)n_in; (void)out_size; (void)ws_size;

  const float* x    = (const float*)d_in[0];
  const float* stat = (const float*)d_in[1];
  const float* hlvs = (const float*)d_in[2];
  const int*   ei   = (const int*)d_in[3];
  const int*   src  = ei;            // edge_index[0]
  const int*   dst  = ei + NE;       // edge_index[1]
  const int*   batch = (const int*)d_in[4];

  const float* cW[4] = {(const float*)d_in[5],  (const float*)d_in[7],
                        (const float*)d_in[9],  (const float*)d_in[11]};
  const float* cB[4] = {(const float*)d_in[6],  (const float*)d_in[8],
                        (const float*)d_in[10], (const float*)d_in[12]};
  const float* nW[4] = {(const float*)d_in[13], (const float*)d_in[15],
                        (const float*)d_in[17], (const float*)d_in[19]};
  const float* nB[4] = {(const float*)d_in[14], (const float*)d_in[16],
                        (const float*)d_in[18], (const float*)d_in[20]};
  const float* hW[4] = {(const float*)d_in[21], (const float*)d_in[23],
                        (const float*)d_in[25], (const float*)d_in[27]};
  const float* hB[4] = {(const float*)d_in[22], (const float*)d_in[24],
                        (const float*)d_in[26], (const float*)d_in[28]};
  const float* eps = (const float*)d_in[29];

  // workspace carve-up
  char* ws = (char*)d_ws;
  size_t off = 0;
  float* h    = (float*)(ws + off); off += align256((size_t)NN * NALL * 4);
  float* agg  = (float*)(ws + off); off += align256((size_t)NN * NALL * 4);
  float* a0   = (float*)(ws + off); off += align256((size_t)NN * DEEP * 4);
  float* a1   = (float*)(ws + off); off += align256((size_t)NN * DEEP * 4);
  float* xd   = (float*)(ws + off); off += align256((size_t)NN * NDYN * 4);
  float* pooled = (float*)(ws + off); off += align256((size_t)NG * NDYN * 4);
  float* hin  = (float*)(ws + off); off += align256((size_t)NG * 68 * 4);

  auto cdiv = [](int a, int b) { return (a + b - 1) / b; };
  auto gemm = [&](const float* A, const float* W, const float* B, float* Cp,
                  int M, int K, int Nf) {
    int Kp = (K + 63) & ~63;
    dim3 grid(cdiv(M / 16, 16), Nf / 64);   // 16 M-tiles (256 rows) per block
    gemm_bias_relu<<<grid, 256, (size_t)64 * Kp * 4, stream>>>(A, W, B, Cp, M, K, Nf);
  };

  init_xd<<<cdiv(NN * NDYN, 256), 256, 0, stream>>>(x, xd);

  for (int it = 0; it < 3; ++it) {
    // GINConv: nn((1+eps)*h + sum_{j->i} h_j)
    build_h<<<cdiv(NN * NALL, 256), 256, 0, stream>>>(xd, stat, hlvs, batch, h);
    zero_f32<<<cdiv(NN * NALL, 256), 256, 0, stream>>>(agg, NN * NALL);
    edge_agg<<<cdiv(NE * NALL, 256), 256, 0, stream>>>(h, src, dst, agg);
    finish_gin<<<cdiv(NN * NALL, 256), 256, 0, stream>>>(h, eps, agg);
    gemm(agg, cW[0], cB[0], a0, NN, NALL, DEEP);
    gemm(a0,  cW[1], cB[1], a1, NN, DEEP, DEEP);
    gemm(a1,  cW[2], cB[2], a0, NN, DEEP, DEEP);
    gemm(a0,  cW[3], cB[3], xd, NN, DEEP, NDYN);
    // node MLP on addstatic(xd)
    build_h<<<cdiv(NN * NALL, 256), 256, 0, stream>>>(xd, stat, hlvs, batch, h);
    gemm(h,  nW[0], nB[0], a0, NN, NALL, DEEP);
    gemm(a0, nW[1], nB[1], a1, NN, DEEP, DEEP);
    gemm(a1, nW[2], nB[2], a0, NN, DEEP, DEEP);
    gemm(a0, nW[3], nB[3], xd, NN, DEEP, NDYN);
  }

  // graph pooling + hlv MLP
  zero_f32<<<cdiv(NG * NDYN, 256), 256, 0, stream>>>(pooled, NG * NDYN);
  pool_sum<<<cdiv(NN * NDYN, 256), 256, 0, stream>>>(xd, batch, pooled);
  build_hin<<<cdiv(NG * 68, 256), 256, 0, stream>>>(hlvs, pooled, hin);
  gemm(hin, hW[0], hB[0], a0, NG, 68,   DEEP);
  gemm(a0,  hW[1], hB[1], a1, NG, DEEP, DEEP);
  gemm(a1,  hW[2], hB[2], a0, NG, DEEP, DEEP);
  final_layer<<<1, NG, 0, stream>>>(a0, hW[3], hB[3], (float*)d_out);
}